// RNN_8031588843828
// MI455X (gfx1250) — compile-verified
//
#include <hip/hip_runtime.h>

// Problem dimensions (compile-time constants from the reference).
#define SEQ   512
#define BATCH 128
#define D_IN  256
#define D_H   1024
#define RANK  128
#define D_OUT 256

typedef __attribute__((ext_vector_type(2))) float v2f;
typedef __attribute__((ext_vector_type(8))) float v8f;

// D(16x16,f32) = A(16x4,f32) x B(4x16,f32) + C  -- CDNA5 fp32 WMMA.
// Fragment layout (ISA 7.12.2): lane = {idx16 = lane&15, half = lane>>4}
//   A: row m = idx16, elements K = 2*half, 2*half+1 -> contiguous float2 of row m
//   B: col n = idx16, elements K = 2*half, 2*half+1 -> contiguous float2 of B row n (B is [N,K])
//   C/D: vgpr r holds (m = r + 8*half, n = idx16)
__device__ __forceinline__ v8f wmma_f32(v2f a, v2f b, v8f c) {
  return __builtin_amdgcn_wmma_f32_16x16x4_f32(false, a, false, b, (short)0, c,
                                               false, false);
}

// ---------------------------------------------------------------------------
// Generic C[M,N] = A[M,K] * B[N,K]^T (+bias).  Block = 256 thr = 8 waves,
// arranged 4 (M) x 2 (N).  Per wave: 2 M-strips x 4 N-tiles = 8 accumulators.
// Block tile: 128 (M) x 128 (N).  6 fragment loads per 8 WMMAs.
// ---------------------------------------------------------------------------
template <bool ADD_BIAS>
__global__ __launch_bounds__(256) void gemm_abt(const float* __restrict__ A,
                                                const float* __restrict__ B,
                                                const float* __restrict__ bias,
                                                float* __restrict__ C,
                                                int M, int N, int K) {
  const int wave   = threadIdx.x >> 5;
  const int lane   = threadIdx.x & 31;
  const int n16    = lane & 15;
  const int kh     = lane >> 4;   // K-pair select for A/B, M-half select for C/D
  const int wave_m = wave >> 1;   // 0..3
  const int wave_n = wave & 1;    // 0..1
  const int m0 = blockIdx.x * 128 + wave_m * 32;   // two 16-row strips: m0, m0+16
  const int n0 = blockIdx.y * 128 + wave_n * 64;   // four 16-col tiles

  const float* aPtr0 = A + (size_t)(m0 + 0  + n16) * K + 2 * kh;
  const float* aPtr1 = A + (size_t)(m0 + 16 + n16) * K + 2 * kh;
  const float* bPtr0 = B + (size_t)(n0 + 0  + n16) * K + 2 * kh;
  const float* bPtr1 = B + (size_t)(n0 + 16 + n16) * K + 2 * kh;
  const float* bPtr2 = B + (size_t)(n0 + 32 + n16) * K + 2 * kh;
  const float* bPtr3 = B + (size_t)(n0 + 48 + n16) * K + 2 * kh;

  v8f acc00 = {}, acc01 = {}, acc02 = {}, acc03 = {};
  v8f acc10 = {}, acc11 = {}, acc12 = {}, acc13 = {};

  for (int k = 0; k < K; k += 8) {
    __builtin_prefetch(aPtr0 + k + 64, 0, 1);  // global_prefetch on streamed A rows
    __builtin_prefetch(aPtr1 + k + 64, 0, 1);
#pragma unroll
    for (int kk = 0; kk < 8; kk += 4) {
      v2f a0 = *(const v2f*)(aPtr0 + k + kk);
      v2f a1 = *(const v2f*)(aPtr1 + k + kk);
      v2f b;
      b = *(const v2f*)(bPtr0 + k + kk);
      acc00 = wmma_f32(a0, b, acc00);
      acc10 = wmma_f32(a1, b, acc10);
      b = *(const v2f*)(bPtr1 + k + kk);
      acc01 = wmma_f32(a0, b, acc01);
      acc11 = wmma_f32(a1, b, acc11);
      b = *(const v2f*)(bPtr2 + k + kk);
      acc02 = wmma_f32(a0, b, acc02);
      acc12 = wmma_f32(a1, b, acc12);
      b = *(const v2f*)(bPtr3 + k + kk);
      acc03 = wmma_f32(a0, b, acc03);
      acc13 = wmma_f32(a1, b, acc13);
    }
  }

  float bv0 = 0.f, bv1 = 0.f, bv2 = 0.f, bv3 = 0.f;
  if (ADD_BIAS) {
    bv0 = bias[n0 + 0  + n16];
    bv1 = bias[n0 + 16 + n16];
    bv2 = bias[n0 + 32 + n16];
    bv3 = bias[n0 + 48 + n16];
  }
#pragma unroll
  for (int r = 0; r < 8; ++r) {
    float* cRow0 = C + (size_t)(m0 + 0  + r + 8 * kh) * N + n0 + n16;
    float* cRow1 = C + (size_t)(m0 + 16 + r + 8 * kh) * N + n0 + n16;
    cRow0[0]  = acc00[r] + bv0;
    cRow0[16] = acc01[r] + bv1;
    cRow0[32] = acc02[r] + bv2;
    cRow0[48] = acc03[r] + bv3;
    cRow1[0]  = acc10[r] + bv0;
    cRow1[16] = acc11[r] + bv1;
    cRow1[32] = acc12[r] + bv2;
    cRow1[48] = acc13[r] + bv3;
  }
}

// ---------------------------------------------------------------------------
// Device-scope spin barrier; one fresh zero-initialized counter per event.
// ---------------------------------------------------------------------------
__device__ __forceinline__ void gridBarrier(unsigned* bar, int idx, int nblk) {
  __syncthreads();
  if (threadIdx.x == 0) {
    __hip_atomic_fetch_add(&bar[idx], 1u, __ATOMIC_RELEASE,
                           __HIP_MEMORY_SCOPE_AGENT);
    while (__hip_atomic_load(&bar[idx], __ATOMIC_ACQUIRE,
                             __HIP_MEMORY_SCOPE_AGENT) < (unsigned)nblk) {
      __builtin_amdgcn_s_sleep(8);
    }
  }
  __syncthreads();
}

// ---------------------------------------------------------------------------
// Persistent recurrence: hidden[] pre-filled with zi; overwritten in place.
//   t=0 : h0 = relu(zi_0 + b_h)
//   t>0 : phase 1: tmpA/tmpB = h_{t-1} @ U^T partials over K-halves
//                  (64 tiles x 2 K-halves -> all 128 waves busy)
//         phase 2: h_t = relu((tmpA+tmpB) @ V^T + b_h + zi_t)
//                  (512 tiles, 4 per wave; partial-sum reduce fused into the
//                   A-fragment load: a = a_lo + a_hi before each WMMA)
// Grid = NBLK blocks x 256 threads (8 waves) -> 128 waves total.
// ---------------------------------------------------------------------------
__global__ __launch_bounds__(256) void rnn_scan(const float* __restrict__ U,
                                                const float* __restrict__ V,
                                                const float* __restrict__ b_h,
                                                float* __restrict__ hidden,
                                                float* __restrict__ tmp,  // 2*[BATCH,RANK]
                                                unsigned* __restrict__ bar,
                                                int nblk) {
  const int wave = threadIdx.x >> 5;
  const int lane = threadIdx.x & 31;
  const int n16  = lane & 15;
  const int kh   = lane >> 4;
  const int gw   = blockIdx.x * 8 + wave;  // 0..127 global wave
  const int tid  = blockIdx.x * blockDim.x + threadIdx.x;
  const int nthr = nblk * 256;

  float* tmpA = tmp;
  float* tmpB = tmp + BATCH * RANK;

  // ---- step 0: h0 = relu(zi_0 + b_h)  (h_{-1} = 0 => zh = b_h) ----
  for (int i = tid; i < BATCH * D_H; i += nthr) {
    float z = hidden[i] + b_h[i & (D_H - 1)];
    hidden[i] = z > 0.f ? z : 0.f;
  }
  gridBarrier(bar, 0, nblk);

  int bidx = 1;
  for (int t = 1; t < SEQ; ++t) {
    const float* hprev = hidden + (size_t)(t - 1) * BATCH * D_H;

    // ---- phase 1: partial tmp = hprev[128,1024] * U[128,1024]^T ----
    {
      const int tileid = gw >> 1;        // 0..63 = 8 (M) x 8 (N) tiles
      const int khalf  = gw & 1;         // K half: [0,512) or [512,1024)
      const int m0 = (tileid >> 3) * 16;
      const int n0 = (tileid & 7) * 16;
      const int kbase = khalf * (D_H / 2);
      const float* aPtr = hprev + (size_t)(m0 + n16) * D_H + kbase + 2 * kh;
      const float* bPtr = U + (size_t)(n0 + n16) * D_H + kbase + 2 * kh;
      float* dst = khalf ? tmpB : tmpA;
      v8f acc = {};
#pragma unroll 8
      for (int k = 0; k < D_H / 2; k += 4) {
        v2f a = *(const v2f*)(aPtr + k);
        v2f b = *(const v2f*)(bPtr + k);
        acc = wmma_f32(a, b, acc);
      }
#pragma unroll
      for (int r = 0; r < 8; ++r)
        dst[(size_t)(m0 + r + 8 * kh) * RANK + n0 + n16] = acc[r];
    }
    gridBarrier(bar, bidx++, nblk);

    // ---- phase 2: h_t = relu((tmpA+tmpB)[128,128] * V[1024,128]^T + b_h + zi_t) ----
    float* hcur = hidden + (size_t)t * BATCH * D_H;
#pragma unroll
    for (int i = 0; i < 4; ++i) {
      const int tile = gw * 4 + i;       // 0..511 = 8 (M) x 64 (N) tiles
      const int m0 = (tile >> 6) * 16;
      const int n0 = (tile & 63) * 16;
      const size_t aOff = (size_t)(m0 + n16) * RANK + 2 * kh;
      const float* aPtrA = tmpA + aOff;
      const float* aPtrB = tmpB + aOff;
      const float* bPtr  = V + (size_t)(n0 + n16) * RANK + 2 * kh;
      v8f acc = {};
#pragma unroll 8
      for (int k = 0; k < RANK; k += 4) {
        v2f a = *(const v2f*)(aPtrA + k) + *(const v2f*)(aPtrB + k);
        v2f b = *(const v2f*)(bPtr + k);
        acc = wmma_f32(a, b, acc);
      }
      const float bj = b_h[n0 + n16];
#pragma unroll
      for (int r = 0; r < 8; ++r) {
        float* p = hcur + (size_t)(m0 + r + 8 * kh) * D_H + n0 + n16;
        float z = *p + acc[r] + bj;      // *p holds zi_t; overwrite with h_t
        *p = z > 0.f ? z : 0.f;
      }
    }
    gridBarrier(bar, bidx++, nblk);
  }
}

__global__ void zero_u32(unsigned* __restrict__ p, int n) {
  int i = blockIdx.x * blockDim.x + threadIdx.x;
  if (i < n) p[i] = 0u;
}

// ---------------------------------------------------------------------------
extern "C" void kernel_launch(void* const* d_in, const int* in_sizes, int n_in,
                              void* d_out, int out_size, void* d_ws,
                              size_t ws_size, hipStream_t stream) {
  const float* x     = (const float*)d_in[0];  // [SEQ,BATCH,D_IN]
  const float* W_in  = (const float*)d_in[1];  // [D_H,D_IN]
  const float* U     = (const float*)d_in[2];  // [RANK,D_H]
  const float* V     = (const float*)d_in[3];  // [D_H,RANK]
  const float* b_h   = (const float*)d_in[4];  // [D_H]
  const float* W_out = (const float*)d_in[5];  // [D_OUT,D_H]
  const float* b_out = (const float*)d_in[6];  // [D_OUT]

  float* hidden = (float*)d_out;                        // [SEQ,BATCH,D_H]
  float* output = hidden + (size_t)SEQ * BATCH * D_H;   // [SEQ,BATCH,D_OUT]

  unsigned* bar = (unsigned*)d_ws;                      // 1024 barrier counters
  float* tmp    = (float*)((char*)d_ws + 4096);         // 2 x [BATCH,RANK] partials

  const int NBAR = 1024;
  zero_u32<<<(NBAR + 255) / 256, 256, 0, stream>>>(bar, NBAR);

  // zi = x @ W_in^T  -> written straight into the hidden output region
  dim3 g1(SEQ * BATCH / 128, D_H / 128);
  gemm_abt<false><<<g1, 256, 0, stream>>>(x, W_in, nullptr, hidden,
                                          SEQ * BATCH, D_H, D_IN);

  const int NBLK = 16;  // small, co-resident grid for the hand-rolled barrier
  rnn_scan<<<NBLK, 256, 0, stream>>>(U, V, b_h, hidden, tmp, bar, NBLK);

  // output = hidden @ W_out^T + b_out
  dim3 g2(SEQ * BATCH / 128, D_OUT / 128);
  gemm_abt<true><<<g2, 256, 0, stream>>>(hidden, W_out, b_out, output,
                                         SEQ * BATCH, D_OUT, D_H);
}